// HardBatchTripletLoss_63556926046454
// MI455X (gfx1250) — compile-verified
//
#include <hip/hip_runtime.h>
#include <math.h>

// ---------------------------------------------------------------------------
// Hard-batch triplet loss, fused Gram-matrix formulation for MI455X (gfx1250).
//
//   d2(i,j) = ||xi||^2 + ||xj||^2 - 2 * dot(xi, xj)
//   loss = mean_i relu( sqrt(max_{pos j} d2) - sqrt(min_{neg j} d2) + 1 )
//
// sqrt is monotone, so max/min reduce on squared distances; one sqrt per row
// at the end. The 17.2 GFLOP Gram matmul runs on the native f32 matrix pipe
// (V_WMMA_F32_16X16X4_F32, exact f32 like the reference). Each wave produces
// a 16x32 output tile with TWO independent WMMA accumulator chains (ILP on
// the matrix pipe), A fragments resident in 64 VGPRs, B panels staged through
// double-buffered LDS so global loads overlap the 64-WMMA compute block.
// ---------------------------------------------------------------------------

typedef float v2f __attribute__((ext_vector_type(2)));
typedef float v8f __attribute__((ext_vector_type(8)));

#define D          128
#define KSTEPS     (D / 4)      // 32 wmma_f32_16x16x4 per 16x16 sub-tile
#define LDS_STRIDE (D + 2)      // 130 % 64 banks == 2 -> conflict-free rows
#define TILE_N     32           // columns per panel (2 accumulators / wave)

// ---------------------------------------------------------------------------
// Kernel 1: per-row squared norms.
// ---------------------------------------------------------------------------
__global__ void sqnorm_kernel(const float* __restrict__ X,
                              float* __restrict__ sqn, int N) {
    int row = blockIdx.x * blockDim.x + threadIdx.x;
    if (row >= N) return;
    const float4* p = (const float4*)(X + (size_t)row * D);
    float s = 0.0f;
#pragma unroll
    for (int i = 0; i < D / 4; ++i) {
        float4 v = p[i];
        s += v.x * v.x + v.y * v.y + v.z * v.z + v.w * v.w;
    }
    sqn[row] = s;
}

// ---------------------------------------------------------------------------
// Kernel 2: fused Gram GEMM + hardest-pos / hardest-neg reduction.
// Grid: N/128 blocks of 256 threads (8 wave32). Wave w owns 16 rows
// (rowBase = (blockIdx*8 + w)*16) and sweeps all N columns in 32-wide panels.
//
// WMMA f32 16x16x4 register layout (ISA 7.12.2):
//   A (16x4): lane L<16 -> M=L,   vgpr0=K0 vgpr1=K1 ; L>=16 -> M=L-16, K2/K3
//   B (4x16): mirrored (lane = N column, lane-half selects K pair)
//   C/D     : lane L -> N=L%16 ; vgpr r -> M = r + 8*(L/16)
// ---------------------------------------------------------------------------
__global__ __launch_bounds__(256)
void triplet_main(const float* __restrict__ X, const int* __restrict__ tgt,
                  const float* __restrict__ sqn, float* __restrict__ slack,
                  int N) {
    __shared__ float lds[2][TILE_N * LDS_STRIDE];   // 2 x 16.6 KB panels

    const int tid  = threadIdx.x;
    const int wave = tid >> 5;
    const int lane = tid & 31;
    const int hi   = lane >> 4;     // lane-half: selects K pair {0,1}/{2,3}
    const int lm   = lane & 15;
    const int rowBase = (blockIdx.x * 8 + wave) * 16;
    const int myRow   = rowBase + lm;

    // --- A fragments: entire 16x128 row strip resident in 64 VGPRs ---
    v2f a[KSTEPS];
#pragma unroll
    for (int s = 0; s < KSTEPS; ++s)
        a[s] = *(const v2f*)(X + (size_t)myRow * D + 4 * s + 2 * hi);

    // --- per-accumulator-row scalars (C/D row index = r + 8*hi) ---
    float sqi[8];
    int   ti[8];
#pragma unroll
    for (int r = 0; r < 8; ++r) {
        sqi[r] = sqn[rowBase + r + 8 * hi];
        ti[r]  = tgt[rowBase + r + 8 * hi];
    }

    float maxpos[8], minneg[8];
#pragma unroll
    for (int r = 0; r < 8; ++r) {
        maxpos[r] = -__builtin_inff();   // self-distance 0 always present
        minneg[r] =  __builtin_inff();
    }

    // coop-staging indices: 32 rows x 32 float4 = 1024 float4, 4 per thread
    const int nTiles = N / TILE_N;

    // ---- preload panel 0 into lds[0] ----
#pragma unroll
    for (int e0 = 0; e0 < 4; ++e0) {
        int e    = tid + e0 * 256;
        int crow = e >> 5;
        int c4   = e & 31;
        float4 v = *(const float4*)(X + (size_t)crow * D + c4 * 4);
        *(float4*)(&lds[0][crow * LDS_STRIDE + c4 * 4]) = v;
    }
    __syncthreads();

    int buf = 0;
    for (int t = 0; t < nTiles; ++t) {
        const int colBase = t * TILE_N;

        // --- issue next panel's global loads BEFORE compute (overlap) ---
        float4 stage[4];
        if (t + 1 < nTiles) {
#pragma unroll
            for (int e0 = 0; e0 < 4; ++e0) {
                int e    = tid + e0 * 256;
                int crow = e >> 5;
                int c4   = e & 31;
                stage[e0] = *(const float4*)(
                    X + (size_t)(colBase + TILE_N + crow) * D + c4 * 4);
            }
            if (t + 2 < nTiles)   // gfx1250 global_prefetch_b8: panel t+2
                __builtin_prefetch(
                    X + (size_t)(colBase + 2 * TILE_N) * D + lane * 16, 0, 0);
        }

        const float* pb = &lds[buf][0];
        const float sqj0 = sqn[colBase + lm];
        const float sqj1 = sqn[colBase + 16 + lm];
        const int   tj0  = tgt[colBase + lm];
        const int   tj1  = tgt[colBase + 16 + lm];

        // --- 64 WMMAs: two independent accumulator chains over K=128 ---
        v8f acc0 = {0.f, 0.f, 0.f, 0.f, 0.f, 0.f, 0.f, 0.f};
        v8f acc1 = {0.f, 0.f, 0.f, 0.f, 0.f, 0.f, 0.f, 0.f};
#pragma unroll
        for (int s = 0; s < KSTEPS; ++s) {
            v2f b0 = *(const v2f*)(pb + lm * LDS_STRIDE + 4 * s + 2 * hi);
            v2f b1 = *(const v2f*)(pb + (lm + 16) * LDS_STRIDE + 4 * s + 2 * hi);
            acc0 = __builtin_amdgcn_wmma_f32_16x16x4_f32(
                false, a[s], false, b0, (short)0, acc0, false, false);
            acc1 = __builtin_amdgcn_wmma_f32_16x16x4_f32(
                false, a[s], false, b1, (short)0, acc1, false, false);
        }

        // --- fused epilogue: masked running max/min on squared distances ---
#pragma unroll
        for (int r = 0; r < 8; ++r) {
            float d2a  = fmaxf(sqi[r] + sqj0 - 2.0f * acc0[r], 0.0f);
            float d2b  = fmaxf(sqi[r] + sqj1 - 2.0f * acc1[r], 0.0f);
            bool  posa = (ti[r] == tj0);
            bool  posb = (ti[r] == tj1);
            maxpos[r] = posa ? fmaxf(maxpos[r], d2a) : maxpos[r];
            minneg[r] = posa ? minneg[r] : fminf(minneg[r], d2a);
            maxpos[r] = posb ? fmaxf(maxpos[r], d2b) : maxpos[r];
            minneg[r] = posb ? minneg[r] : fminf(minneg[r], d2b);
        }

        // --- store staged panel into the other buffer, single barrier ---
        if (t + 1 < nTiles) {
#pragma unroll
            for (int e0 = 0; e0 < 4; ++e0) {
                int e    = tid + e0 * 256;
                int crow = e >> 5;
                int c4   = e & 31;
                *(float4*)(&lds[buf ^ 1][crow * LDS_STRIDE + c4 * 4]) = stage[e0];
            }
        }
        __syncthreads();
        buf ^= 1;
    }

    // butterfly reduce across the 16 lanes sharing each row (per lane-half)
#pragma unroll
    for (int m = 1; m < 16; m <<= 1) {
#pragma unroll
        for (int r = 0; r < 8; ++r) {
            maxpos[r] = fmaxf(maxpos[r], __shfl_xor(maxpos[r], m, 32));
            minneg[r] = fminf(minneg[r], __shfl_xor(minneg[r], m, 32));
        }
    }

    if (lm == 0) {
#pragma unroll
        for (int r = 0; r < 8; ++r) {
            int   row = rowBase + r + 8 * hi;
            float hp  = sqrtf(fmaxf(maxpos[r], 0.0f));
            float hn  = sqrtf(minneg[r]);               // +inf -> slack 0
            slack[row] = fmaxf(hp - hn + 1.0f, 0.0f);   // MARGIN = 1.0
        }
    }
}

// ---------------------------------------------------------------------------
// Kernel 3: deterministic single-block mean reduction (no float atomics).
// ---------------------------------------------------------------------------
__global__ void reduce_kernel(const float* __restrict__ slack,
                              float* __restrict__ out, int N) {
    __shared__ float sm[256];
    int tid = threadIdx.x;
    float s = 0.0f;
    for (int i = tid; i < N; i += 256) s += slack[i];
    sm[tid] = s;
    __syncthreads();
    for (int k = 128; k > 0; k >>= 1) {
        if (tid < k) sm[tid] += sm[tid + k];
        __syncthreads();
    }
    if (tid == 0) out[0] = sm[0] / (float)N;
}

// ---------------------------------------------------------------------------
extern "C" void kernel_launch(void* const* d_in, const int* in_sizes, int n_in,
                              void* d_out, int out_size, void* d_ws,
                              size_t ws_size, hipStream_t stream) {
    const float* X   = (const float*)d_in[0];   // [N, 128] f32
    const int*   tgt = (const int*)d_in[1];     // [N] (harness int ABI)
    const int    N   = in_sizes[1];             // 8192

    float* ws    = (float*)d_ws;
    float* sqn   = ws;          // N floats
    float* slack = ws + N;      // N floats
    float* out   = (float*)d_out;

    sqnorm_kernel<<<(N + 255) / 256, 256, 0, stream>>>(X, sqn, N);
    triplet_main<<<N / 128, 256, 0, stream>>>(X, tgt, sqn, slack, N);
    reduce_kernel<<<1, 256, 0, stream>>>(slack, out, N);
}